// PerformerAttentionCore_49237505081856
// MI455X (gfx1250) — compile-verified
//
#include <hip/hip_runtime.h>

// ---------------------------------------------------------------------------
// Performer (FAVOR+) causal linear attention for MI455X (gfx1250).
// B=2, H=16, N=4096, D=64, M=256 features.  All f32.
//  - every contraction via V_WMMA_F32_16X16X4_F32
//  - LDS-resident state S[256][64] + z[256]
//  - double-buffered chunk staging with GLOBAL_LOAD_ASYNC_TO_LDS_B128
//    overlapped across the sequential chunk recurrence (ASYNCcnt pipelining)
// ---------------------------------------------------------------------------

typedef __attribute__((ext_vector_type(2))) float v2f;
typedef __attribute__((ext_vector_type(8))) float v8f;

#define HD      64      // head dim
#define MF      256     // num features
#define SEQ_N   4096
#define CHUNK   32
#define NCHUNK  (SEQ_N / CHUNK)
#define BH      32      // B*H

// padded LDS strides (bank-conflict avoidance; QS mult-of-4 for b128 targets)
#define OS      66      // omega row stride
#define QS      68      // q/k chunk row stride (16B-aligned rows for async b128)
#define PS      258     // phi row stride
#define GS      34      // G row stride

__device__ __forceinline__ v8f wmma_f32(v2f a, v2f b, v8f c) {
  // D = A(16x4) * B(4x16) + C(16x16), all f32
  return __builtin_amdgcn_wmma_f32_16x16x4_f32(
      /*neg_a=*/false, a, /*neg_b=*/false, b,
      /*c_mod=*/(short)0, c, /*reuse_a=*/false, /*reuse_b=*/false);
}

// async memory->LDS copy of one b128 (4 floats) per lane, GV addressing
__device__ __forceinline__ void async_b128(const float* gsrc, const float* ldst) {
  unsigned loff = (unsigned)(uintptr_t)ldst;  // low 32 bits = LDS byte offset
  asm volatile("global_load_async_to_lds_b128 %0, %1, off"
               :: "v"(loff), "v"(gsrc)
               : "memory");
}

// issue the 6 async b128 transfers this thread owns for one chunk (q,k,v)
__device__ __forceinline__ void async_copy_chunk(const float* gq, const float* gk,
                                                 const float* gv, float* bq,
                                                 float* bk, float* bv, int c0,
                                                 int tid) {
#pragma unroll
  for (int it = 0; it < 2; ++it) {
    const int idx = tid + (it << 8);   // 0..511
    const int r = idx >> 4;            // row 0..31
    const int dq = (idx & 15) << 2;    // dword-quad 0..60
    const size_t g = (size_t)(c0 + r) * HD + dq;
    async_b128(gq + g, &bq[r * QS + dq]);
    async_b128(gk + g, &bk[r * QS + dq]);
    async_b128(gv + g, &bv[r * HD + dq]);
  }
}

// ---------------------------------------------------------------------------
// Kernel 1: per-block partial maxima of log_phi for keys (global max prepass)
// ---------------------------------------------------------------------------
__global__ __launch_bounds__(256)
void performer_keymax_partial(const float* __restrict__ kin,
                              const float* __restrict__ omega,
                              float* __restrict__ partials) {
  __shared__ float sOm[MF * HD];
  __shared__ float sred[8];
  for (int i = threadIdx.x; i < MF * HD; i += 256) sOm[i] = omega[i];
  __syncthreads();

  const float scale = 0.35355339059327379f;  // 64^-0.25
  const int row = blockIdx.x * 256 + threadIdx.x;  // < 131072
  const float* kr = kin + (size_t)row * HD;

  float x[HD];
  float nrm = 0.f;
#pragma unroll
  for (int d = 0; d < HD; ++d) {
    float t = kr[d] * scale;
    x[d] = t;
    nrm += t * t;
  }
  nrm *= 0.5f;

  float mx = -3.4e38f;
  for (int m = 0; m < MF; ++m) {
    const float* om = &sOm[m * HD];
    float dp = 0.f;
#pragma unroll
    for (int d = 0; d < HD; ++d) dp = fmaf(x[d], om[d], dp);
    mx = fmaxf(mx, dp - nrm);
  }
#pragma unroll
  for (int off = 16; off > 0; off >>= 1) mx = fmaxf(mx, __shfl_xor(mx, off, 32));
  if ((threadIdx.x & 31) == 0) sred[threadIdx.x >> 5] = mx;
  __syncthreads();
  if (threadIdx.x == 0) {
    float m2 = sred[0];
#pragma unroll
    for (int i = 1; i < 8; ++i) m2 = fmaxf(m2, sred[i]);
    partials[blockIdx.x] = m2;
  }
}

// ---------------------------------------------------------------------------
// Kernel 2: reduce 512 partials -> gmax scalar
// ---------------------------------------------------------------------------
__global__ __launch_bounds__(256)
void performer_keymax_reduce(const float* __restrict__ partials,
                             float* __restrict__ gmax) {
  __shared__ float sred[8];
  float mx = -3.4e38f;
  for (int i = threadIdx.x; i < 512; i += 256) mx = fmaxf(mx, partials[i]);
#pragma unroll
  for (int off = 16; off > 0; off >>= 1) mx = fmaxf(mx, __shfl_xor(mx, off, 32));
  if ((threadIdx.x & 31) == 0) sred[threadIdx.x >> 5] = mx;
  __syncthreads();
  if (threadIdx.x == 0) {
    float m2 = sred[0];
#pragma unroll
    for (int i = 1; i < 8; ++i) m2 = fmaxf(m2, sred[i]);
    gmax[0] = m2;
  }
}

// ---------------------------------------------------------------------------
// Kernel 3: chunked causal linear-attention scan.  One block per (b,h).
// ---------------------------------------------------------------------------
__global__ __launch_bounds__(256)
void performer_scan(const float* __restrict__ qin,
                    const float* __restrict__ kin,
                    const float* __restrict__ vin,
                    const float* __restrict__ omega,
                    const float* __restrict__ gmaxp,
                    float* __restrict__ out) {
  __shared__ float sOm[MF * OS];          // scale * omega [m][d], padded
  __shared__ float sS[MF * HD];           // state S [m][d]
  __shared__ float sZ[MF];                // state z [m]
  __shared__ float sQ[2][CHUNK * QS];     // raw q chunk (double buffered)
  __shared__ float sK[2][CHUNK * QS];     // raw k chunk
  __shared__ float sV[2][CHUNK * HD];     // v chunk
  __shared__ float sPhiQ[CHUNK * PS];     // phi(q) [i][m]
  __shared__ float sPhiK[CHUNK * PS];     // phi(k) [i][m]
  __shared__ float sG[CHUNK * GS];        // phiQ . phiK^T [i][j]
  __shared__ float sNum[CHUNK * HD];      // numerator [i][d]
  __shared__ float sDen[CHUNK];
  __shared__ float sQmax[CHUNK];          // rowwise max of q projection
  __shared__ float sKn[CHUNK];            // 0.5*scale^2*||k||^2 per row

  const int tid = threadIdx.x;
  const int lane = tid & 31;
  const int wv = tid >> 5;               // 8 waves
  const int l16 = lane & 15;
  const int khalf = (lane >> 4) << 1;    // 0 or 2: K-half select for A/B ops
  const int rofs = (lane >> 4) << 3;     // 0 or 8: row offset for C/D tiles
  const int bh = blockIdx.x;
  const float scale = 0.35355339059327379f;   // 64^-0.25
  const float hsc2 = 0.0625f;                 // 0.5 * scale^2
  const float gmax = gmaxp[0];

  const float* qb = qin + (size_t)bh * SEQ_N * HD;
  const float* kb = kin + (size_t)bh * SEQ_N * HD;
  const float* vb = vin + (size_t)bh * SEQ_N * HD;
  float* ob = out + (size_t)bh * SEQ_N * HD;

  // init omega (pre-scaled: proj = q . (scale*omega)) + state
  for (int i = tid; i < MF * HD; i += 256) {
    int m = i >> 6, d = i & 63;
    sOm[m * OS + d] = omega[i] * scale;
  }
  for (int i = tid; i < MF * HD; i += 256) sS[i] = 0.f;
  if (tid < MF) sZ[tid] = 0.f;

  // prologue: kick off chunk 0 into buffer 0 (overlaps with init above)
  async_copy_chunk(qb, kb, vb, sQ[0], sK[0], sV[0], 0, tid);
  __syncthreads();

  for (int ci = 0; ci < NCHUNK; ++ci) {
    const int c0 = ci * CHUNK;
    const int cur = ci & 1;
    const float* cQ = sQ[cur];
    const float* cK = sK[cur];
    const float* cV = sV[cur];

    // issue next chunk into the other buffer, then wait only for the
    // previous chunk's 6 transfers (async loads complete in order).
    if (ci + 1 < NCHUNK) {
      async_copy_chunk(qb, kb, vb, sQ[cur ^ 1], sK[cur ^ 1], sV[cur ^ 1],
                       c0 + CHUNK, tid);
      asm volatile("s_wait_asynccnt 0x6" ::: "memory");
    } else {
      asm volatile("s_wait_asynccnt 0x0" ::: "memory");
    }
    __syncthreads();  // current chunk resident in LDS for all waves

    // ---- row norms for k (q's norm cancels in its row-max) ---------------
    if (tid < CHUNK) {
      float ak = 0.f;
      for (int d = 0; d < HD; ++d) {
        float y = cK[tid * QS + d];
        ak = fmaf(y, y, ak);
      }
      sKn[tid] = hsc2 * ak;
    }
    __syncthreads();

    // ---- projections  proj = chunk[32x64] x (scale*omega)^T[64x256] ------
    // 64 output tiles (32 for Q, 32 for K); 8 per wave.
    for (int t = wv * 8; t < wv * 8 + 8; ++t) {
      const int mats = t >> 5;                 // 0 = Q, 1 = K
      const int loc = t & 31;
      const int row0 = (loc >> 4) * 16;
      const int col0 = (loc & 15) * 16;
      const float* src = mats ? cK : cQ;
      float* dst = mats ? sPhiK : sPhiQ;
      v8f acc = {};
      for (int kk = 0; kk < HD; kk += 4) {
        const int kb2 = kk + khalf;
        v2f a = *(const v2f*)&src[(row0 + l16) * QS + kb2];
        v2f b = *(const v2f*)&sOm[(col0 + l16) * OS + kb2];
        acc = wmma_f32(a, b, acc);
      }
#pragma unroll
      for (int j = 0; j < 8; ++j)
        dst[(row0 + rofs + j) * PS + col0 + l16] = acc[j];
    }
    __syncthreads();

    // ---- stabilized exp feature map --------------------------------------
    for (int r = wv * 4; r < wv * 4 + 4; ++r) {
      float mx = -3.4e38f;
      for (int m = lane; m < MF; m += 32) mx = fmaxf(mx, sPhiQ[r * PS + m]);
#pragma unroll
      for (int off = 16; off > 0; off >>= 1)
        mx = fmaxf(mx, __shfl_xor(mx, off, 32));
      if (lane == 0) sQmax[r] = mx;
    }
    __syncthreads();
    for (int i = tid; i < CHUNK * MF; i += 256) {
      int r = i >> 8, m = i & 255;
      float pq = sPhiQ[r * PS + m];
      sPhiQ[r * PS + m] = __expf(pq - sQmax[r]) * 0.0625f + 1e-4f;
      float pk = sPhiK[r * PS + m];
      sPhiK[r * PS + m] = __expf(pk - sKn[r] - gmax) * 0.0625f + 1e-4f;
    }
    __syncthreads();

    // ---- inter-chunk terms: den0 = phiQ.z ; Num0 = phiQ x S --------------
    for (int r = wv * 4; r < wv * 4 + 4; ++r) {
      float s = 0.f;
      for (int m = lane; m < MF; m += 32)
        s = fmaf(sPhiQ[r * PS + m], sZ[m], s);
#pragma unroll
      for (int off = 16; off > 0; off >>= 1) s += __shfl_xor(s, off, 32);
      if (lane == 0) sDen[r] = s + 1e-6f;
    }
    {
      const int row0 = (wv >> 2) * 16;
      const int col0 = (wv & 3) * 16;
      v8f acc = {};
      for (int kk = 0; kk < MF; kk += 4) {
        const int kb2 = kk + khalf;
        v2f a = *(const v2f*)&sPhiQ[(row0 + l16) * PS + kb2];
        v2f b;
        b.x = sS[kb2 * HD + col0 + l16];
        b.y = sS[(kb2 + 1) * HD + col0 + l16];
        acc = wmma_f32(a, b, acc);
      }
#pragma unroll
      for (int j = 0; j < 8; ++j)
        sNum[(row0 + rofs + j) * HD + col0 + l16] = acc[j];
    }
    __syncthreads();

    // ---- G = phiQ x phiK^T  [32x32]; 4 tiles on waves 0-3 ----------------
    if (wv < 4) {
      const int row0 = (wv >> 1) * 16;
      const int col0 = (wv & 1) * 16;
      v8f acc = {};
      for (int kk = 0; kk < MF; kk += 4) {
        const int kb2 = kk + khalf;
        v2f a = *(const v2f*)&sPhiQ[(row0 + l16) * PS + kb2];
        v2f b = *(const v2f*)&sPhiK[(col0 + l16) * PS + kb2];
        acc = wmma_f32(a, b, acc);
      }
#pragma unroll
      for (int j = 0; j < 8; ++j)
        sG[(row0 + rofs + j) * GS + col0 + l16] = acc[j];
    }
    __syncthreads();

    // ---- causal mask + rowsum into den -----------------------------------
    if (tid < CHUNK) {
      float rs = 0.f;
      for (int j = 0; j <= tid; ++j) rs += sG[tid * GS + j];
      for (int j = tid + 1; j < CHUNK; ++j) sG[tid * GS + j] = 0.f;
      sDen[tid] += rs;
    }
    __syncthreads();

    // ---- Num += tril(G)[32x32] x V[32x64]; 8 tiles -----------------------
    {
      const int row0 = (wv >> 2) * 16;
      const int col0 = (wv & 3) * 16;
      v8f acc;
#pragma unroll
      for (int j = 0; j < 8; ++j)
        acc[j] = sNum[(row0 + rofs + j) * HD + col0 + l16];
      for (int kk = 0; kk < CHUNK; kk += 4) {
        const int kb2 = kk + khalf;
        v2f a = *(const v2f*)&sG[(row0 + l16) * GS + kb2];
        v2f b;
        b.x = cV[kb2 * HD + col0 + l16];
        b.y = cV[(kb2 + 1) * HD + col0 + l16];
        acc = wmma_f32(a, b, acc);
      }
#pragma unroll
      for (int j = 0; j < 8; ++j)
        sNum[(row0 + rofs + j) * HD + col0 + l16] = acc[j];
    }
    __syncthreads();

    // ---- write output  out = Num / den -----------------------------------
    for (int i = tid; i < CHUNK * HD; i += 256) {
      int r = i >> 6, d = i & 63;
      ob[(size_t)(c0 + r) * HD + d] = sNum[r * HD + d] / sDen[r];
    }

    // ---- state update  S += phiK^T[256x32] x V[32x64]; z += colsum(phiK) -
    for (int s8 = 0; s8 < 8; ++s8) {
      const int t = wv * 8 + s8;
      const int row0 = (t >> 2) * 16;   // M dim
      const int col0 = (t & 3) * 16;    // D dim
      v8f acc;
#pragma unroll
      for (int j = 0; j < 8; ++j)
        acc[j] = sS[(row0 + rofs + j) * HD + col0 + l16];
      for (int kk = 0; kk < CHUNK; kk += 4) {
        const int kb2 = kk + khalf;
        v2f a;
        a.x = sPhiK[kb2 * PS + row0 + l16];        // phiK^T[m][j]
        a.y = sPhiK[(kb2 + 1) * PS + row0 + l16];
        v2f b;
        b.x = cV[kb2 * HD + col0 + l16];
        b.y = cV[(kb2 + 1) * HD + col0 + l16];
        acc = wmma_f32(a, b, acc);
      }
#pragma unroll
      for (int j = 0; j < 8; ++j)
        sS[(row0 + rofs + j) * HD + col0 + l16] = acc[j];
    }
    {
      float zz = 0.f;
      for (int j = 0; j < CHUNK; ++j) zz += sPhiK[j * PS + tid];
      sZ[tid] += zz;
    }
    __syncthreads();
  }
}

// ---------------------------------------------------------------------------
// Launch: inputs are q, k, v, omega (all f32). ws: [0]=gmax, [1..512]=partials
// ---------------------------------------------------------------------------
extern "C" void kernel_launch(void* const* d_in, const int* in_sizes, int n_in,
                              void* d_out, int out_size, void* d_ws,
                              size_t ws_size, hipStream_t stream) {
  (void)in_sizes; (void)n_in; (void)out_size; (void)ws_size;
  const float* q = (const float*)d_in[0];
  const float* k = (const float*)d_in[1];
  const float* v = (const float*)d_in[2];
  const float* omega = (const float*)d_in[3];
  float* out = (float*)d_out;
  float* ws = (float*)d_ws;

  performer_keymax_partial<<<512, 256, 0, stream>>>(k, omega, ws + 1);
  performer_keymax_reduce<<<1, 256, 0, stream>>>(ws + 1, ws);
  performer_scan<<<BH, 256, 0, stream>>>(q, k, v, omega, ws, out);
}